// DocumentModel_81192061763867
// MI455X (gfx1250) — compile-verified
//
#include <hip/hip_runtime.h>

#define EMB 100
#define BOT 5

typedef __attribute__((ext_vector_type(2))) float v2f;
typedef __attribute__((ext_vector_type(8))) float v8f;

// ---------------- kernel 1: zero the workspace (tf counts, denom, doc) ----
__global__ void zero_ws_kernel(unsigned* __restrict__ ws, size_t nwords) {
  size_t i = (size_t)blockIdx.x * blockDim.x + threadIdx.x;
  const size_t stride = (size_t)gridDim.x * blockDim.x;
  for (; i < nwords; i += stride) ws[i] = 0u;
}

// ---------------- kernel 2: per-row token histogram + idf row-sum ---------
// grid: (ceil(S/256), B), block: 256
__global__ void __launch_bounds__(256)
hist_kernel(const int* __restrict__ x, const float* __restrict__ idf,
            unsigned* __restrict__ tf, float* __restrict__ denom,
            int S, int V) {
  __shared__ float red[256];
  const int b = blockIdx.y;
  const int s = blockIdx.x * 256 + threadIdx.x;
  float v = 0.0f;
  if (s < S) {
    const int tok = x[(size_t)b * S + s];
    atomicAdd(&tf[(size_t)b * V + tok], 1u);
    v = idf[tok];
  }
  red[threadIdx.x] = v;
  __syncthreads();
  for (int off = 128; off > 0; off >>= 1) {
    if (threadIdx.x < off) red[threadIdx.x] += red[threadIdx.x + off];
    __syncthreads();
  }
  if (threadIdx.x == 0) atomicAdd(&denom[b], red[0]);
}

// ---------------- kernel 3: doc[b,:] = sum_s w(b,s) * emb[x[b,s],:] -------
// grid: B blocks, block: 256. Threads cooperatively stage (token, weight)
// tiles in LDS; threads 0..EMB-1 each own one embedding dim.
__global__ void __launch_bounds__(256)
doc_kernel(const int* __restrict__ x, const float* __restrict__ idf,
           const float* __restrict__ emb, const unsigned* __restrict__ tf,
           const float* __restrict__ denom, float* __restrict__ doc,
           int S, int V) {
  __shared__ float w_s[256];
  __shared__ int   t_s[256];
  const int b = blockIdx.x;
  const int e = threadIdx.x;
  const float inv_denom = 1.0f / denom[b];
  float acc = 0.0f;
  for (int s0 = 0; s0 < S; s0 += 256) {
    const int s = s0 + threadIdx.x;
    // prefetch next token tile (global_prefetch_b8) while this one computes
    if (s + 256 < S) __builtin_prefetch(&x[(size_t)b * S + s + 256], 0, 0);
    int tok = 0;
    float w = 0.0f;
    if (s < S) {
      tok = x[(size_t)b * S + s];
      w = (float)tf[(size_t)b * V + tok] * idf[tok] * inv_denom;
    }
    t_s[threadIdx.x] = tok;
    w_s[threadIdx.x] = w;
    __syncthreads();
    if (e < EMB) {
#pragma unroll 4
      for (int t = 0; t < 256; ++t)
        acc = fmaf(w_s[t], emb[(size_t)t_s[t] * EMB + e], acc);
    }
    __syncthreads();
  }
  if (e < EMB) doc[(size_t)b * EMB + e] = acc;
}

// ---------------- kernel 4: MLP via V_WMMA_F32_16X16X4_F32 ----------------
// out = (doc @ fc1_w^T + fc1_b) @ fc2_w^T + fc2_b
// One block, 8 waves; wave m owns output rows 16m..16m+15.
// 16x16x4 f32 layouts (ISA 7.12.2): per lane (l=lane&15, h=lane>>4):
//   A vgpr v  -> A[M=l][K=2h+v]
//   B vgpr v  -> B[K=2h+v][N=l]
//   C/D vgpr r-> D[M=r+8h][N=l]
// Padded columns/rows are produced branchlessly: clamp the index so the
// load is always in-bounds, then multiply by a 0/1 mask (no exec toggling).
__global__ void __launch_bounds__(256)
mlp_wmma_kernel(const float* __restrict__ doc,
                const float* __restrict__ fc1w, const float* __restrict__ fc1b,
                const float* __restrict__ fc2w, const float* __restrict__ fc2b,
                float* __restrict__ out, int B) {
  __shared__ float h_lds[128][17];  // padded: conflict-free column reads
  const int tid  = threadIdx.x;
  const int wave = tid >> 5;
  const int lane = tid & 31;
  const int half = lane >> 4;
  const int l    = lane & 15;
  const int mrow = wave * 16;
  if (mrow >= B) return;  // wave-uniform: remaining waves keep EXEC all-ones

  // ---- GEMM1: h[128,16] = doc[128,100] x fc1w^T[100,16pad] ----
  const int   lc = (l < BOT) ? l : 0;      // clamped bottleneck column
  const float ml = (l < BOT) ? 1.0f : 0.0f;
  v8f c = {};
#pragma unroll
  for (int kk = 0; kk < EMB / 4; ++kk) {  // 25 K-steps of 4
    const int k0 = kk * 4 + half * 2;
    v2f a, bb;
    a[0] = doc[(size_t)(mrow + l) * EMB + k0];
    a[1] = doc[(size_t)(mrow + l) * EMB + k0 + 1];
    bb[0] = ml * fc1w[(size_t)lc * EMB + k0];
    bb[1] = ml * fc1w[(size_t)lc * EMB + k0 + 1];
    c = __builtin_amdgcn_wmma_f32_16x16x4_f32(false, a, false, bb,
                                              (short)0, c, false, false);
  }
  // bias on valid bottleneck cols; cols >= BOT stay exactly zero
  const float b1 = ml * ((l < BOT) ? fc1b[lc] : 0.0f);
#pragma unroll
  for (int r = 0; r < 8; ++r) {
    h_lds[mrow + r + 8 * half][l] = c[r] + b1;
  }
  __syncthreads();

  // ---- GEMM2: out[128,100] = h[128,8pad] x fc2w^T[8pad,112pad] ----
#pragma unroll
  for (int nt = 0; nt < 7; ++nt) {  // 7 N-tiles cover 100 (pad to 112)
    const int   n  = nt * 16 + l;
    const int   nc = (n < EMB) ? n : 0;
    const float mn = (n < EMB) ? 1.0f : 0.0f;
    v8f d = {};
#pragma unroll
    for (int kk = 0; kk < 2; ++kk) {  // K = 8 (BOT=5 zero-padded)
      const int k0 = kk * 4 + half * 2;
      const int   k0c = (k0 < BOT) ? k0 : 0;
      const float m0  = (k0 < BOT) ? mn : 0.0f;
      const int   k1c = (k0 + 1 < BOT) ? (k0 + 1) : 0;
      const float m1  = (k0 + 1 < BOT) ? mn : 0.0f;
      v2f a, bb;
      a[0] = h_lds[mrow + l][k0];
      a[1] = h_lds[mrow + l][k0 + 1];
      bb[0] = m0 * fc2w[(size_t)nc * BOT + k0c];
      bb[1] = m1 * fc2w[(size_t)nc * BOT + k1c];
      d = __builtin_amdgcn_wmma_f32_16x16x4_f32(false, a, false, bb,
                                                (short)0, d, false, false);
    }
    if (n < EMB) {
      const float b2 = fc2b[n];
#pragma unroll
      for (int r = 0; r < 8; ++r)
        out[(size_t)(mrow + r + 8 * half) * EMB + n] = d[r] + b2;
    }
  }
}

// --------------------------------------------------------------------------
extern "C" void kernel_launch(void* const* d_in, const int* in_sizes, int n_in,
                              void* d_out, int out_size, void* d_ws, size_t ws_size,
                              hipStream_t stream) {
  const int*   x    = (const int*)d_in[0];    // [B,S] token ids (int32)
  const float* we   = (const float*)d_in[1];  // [V,EMB]
  const float* idf  = (const float*)d_in[2];  // [V]
  const float* fc1w = (const float*)d_in[3];  // [BOT,EMB]
  const float* fc1b = (const float*)d_in[4];  // [BOT]
  const float* fc2w = (const float*)d_in[5];  // [EMB,BOT]
  const float* fc2b = (const float*)d_in[6];  // [EMB]
  float* out = (float*)d_out;

  const int V = in_sizes[2];
  const int B = out_size / EMB;        // 128
  const int S = in_sizes[0] / B;       // 2048

  // workspace layout: [tf: B*V u32][denom: B f32][doc: B*EMB f32]
  unsigned* tf    = (unsigned*)d_ws;
  float*    denom = (float*)(tf + (size_t)B * V);
  float*    doc   = denom + B;
  const size_t nwords = (size_t)B * V + (size_t)B + (size_t)B * EMB;

  zero_ws_kernel<<<2048, 256, 0, stream>>>((unsigned*)d_ws, nwords);

  dim3 hgrid((S + 255) / 256, B);
  hist_kernel<<<hgrid, 256, 0, stream>>>(x, idf, tf, denom, S, V);

  doc_kernel<<<B, 256, 0, stream>>>(x, idf, we, tf, denom, doc, S, V);

  mlp_wmma_kernel<<<1, 256, 0, stream>>>(doc, fc1w, fc1b, fc2w, fc2b, out, B);
}